// TensorNet_33208687132874
// MI455X (gfx1250) — compile-verified
//
#include <hip/hip_runtime.h>
#include <hip/hip_bf16.h>

typedef __attribute__((ext_vector_type(2))) float v2f;
typedef __attribute__((ext_vector_type(4))) float v4f;
typedef __attribute__((ext_vector_type(8))) float v8f;

namespace {
constexpr int L   = 128;                 // sequence steps
constexpr int D   = 64;                  // embedding dim (contracted)
constexpr int O   = 32;                  // output dim
constexpr int CHI = 16;                  // bond dim
constexpr int E   = CHI * CHI;           // 256 floats per (chi,chi) matrix
constexpr size_t DSTRIDE = (size_t)L * O * E;   // mps stride over d = 1,048,576 floats
}

#if __has_builtin(__builtin_amdgcn_global_load_async_to_lds_b128) && \
    __has_builtin(__builtin_amdgcn_s_wait_asynccnt)
#define TN_USE_ASYNC_LDS 1
// The builtin wants pointers to a 16-byte int vector, AS(1) source / AS(3) dest
// (clang prints these LangASes as __device__ / __shared__ in HIP).
typedef int v4i_vs __attribute__((__vector_size__(4 * sizeof(int))));
typedef __attribute__((address_space(1))) v4i_vs g4i_t;   // global int4
typedef __attribute__((address_space(3))) v4i_vs l4i_t;   // LDS int4
#else
#define TN_USE_ASYNC_LDS 0
#endif

// One wave32 per (l,o) pair. mps slice for the pair (64 KB) is staged into LDS
// once (async copy, ASYNCcnt-tracked, bypassing VGPRs when available); the
// whole 128-step recurrence then runs out of LDS with f32 WMMA.
__global__ __launch_bounds__(32, 1) void tensornet_mps_scan(
    const float* __restrict__ x,    // (L, D)
    const float* __restrict__ mps,  // (D, L, O, CHI, CHI)
    float* __restrict__ out)        // (L, O) flattened
{
  __shared__ float Ms[D * E];       // 64 KB: mps[:, l, o, :, :], layout [d][i*16+j]
  __shared__ float matT[16 * E];    // 16 KB: the 16 per-step matrices of one t-tile
  __shared__ float ev[E + 16];      // env (16x16), padded

  const int lane = threadIdx.x;     // 0..31, single wave per workgroup
  const int p    = blockIdx.x;      // pair index: l = p>>5, o = p&31
  const int col  = lane & 15;       // N / M index held by this lane
  const int hi   = lane >> 4;       // lane half
  const int dsel = hi << 1;         // A/B K-select: lanes 0-15 -> K0/K1, 16-31 -> K2/K3
  const int mrow = hi << 3;         // C/D row base: 0 or 8

  // ---- Stage Ms = mps[:, l, o, :, :] into LDS (mps read exactly once) ----
  const float* mbase = mps + (size_t)p * E;
#if TN_USE_ASYNC_LDS
  // 128 x GLOBAL_LOAD_ASYNC_TO_LDS_B128: 32 lanes x 16B per op, no VGPR landing.
  // Hardware backpressures the 6-bit ASYNCcnt; one wait drains everything.
  for (int d = 0; d < D; ++d) {
    const float* src = mbase + (size_t)d * DSTRIDE;
    float* dst = &Ms[d * E];
    __builtin_amdgcn_global_load_async_to_lds_b128(
        (g4i_t*)(src + lane * 4),       (l4i_t*)(dst + lane * 4),       0, 0);
    __builtin_amdgcn_global_load_async_to_lds_b128(
        (g4i_t*)(src + 128 + lane * 4), (l4i_t*)(dst + 128 + lane * 4), 0, 0);
  }
  __builtin_amdgcn_s_wait_asynccnt(0);
  asm volatile("" ::: "memory");    // pin compile-time ordering vs. later ds reads
#else
  for (int d = 0; d < D; ++d) {
    const float* src = mbase + (size_t)d * DSTRIDE;
    if (d + 4 < D) __builtin_prefetch(src + 4 * DSTRIDE, 0, 0);
    *(v4f*)&Ms[d * E + lane * 8]     = *(const v4f*)&src[lane * 8];
    *(v4f*)&Ms[d * E + lane * 8 + 4] = *(const v4f*)&src[lane * 8 + 4];
  }
#endif
  __builtin_amdgcn_s_barrier_signal(-1);  // single-wave WG: no-op barrier
  __builtin_amdgcn_s_barrier_wait(-1);

  for (int tt = 0; tt < L / 16; ++tt) {   // 8 t-tiles of 16 steps
    const int t0 = tt * 16;

    // A operands for X tile: A[m,k] = x[t0+m, 4*kc + k]; lane holds row m=col.
    // x is 32 KB and shared by all 4096 blocks -> L2-resident global reads.
    v2f A[16];
#pragma unroll
    for (int kc = 0; kc < 16; ++kc)
      A[kc] = *(const v2f*)&x[(t0 + col) * D + kc * 4 + dsel];

    // mats GEMM: matT(16 x 256) = X_tile(16x64) @ Ms(64x256), K accumulated in WMMA
    for (int n = 0; n < 16; ++n) {        // N-tiles = rows i of each 16x16 mat
      v8f c = {};
#pragma unroll
      for (int kc = 0; kc < 16; ++kc) {
        v2f b;                            // B[k,j] = Ms[4kc+k][n*16+j], lane holds j=col
        b.x = Ms[(kc * 4 + dsel) * E + n * 16 + col];
        b.y = Ms[(kc * 4 + dsel + 1) * E + n * 16 + col];
        c = __builtin_amdgcn_wmma_f32_16x16x4_f32(false, A[kc], false, b,
                                                  (short)0, c, false, false);
      }
      // C[m,j] -> matT[m][n][j]; lane holds j=col, rows mrow..mrow+7
#pragma unroll
      for (int v = 0; v < 8; ++v)
        matT[(mrow + v) * E + n * 16 + col] = c[v];
    }

    // ---- sequential chain over the 16 steps of this tile ----
    int mstart = 0;
    if (tt == 0) {                        // env0 = mat_{t=0} (no sigmoid on step 0)
#pragma unroll
      for (int e = lane; e < E; e += 32) ev[e] = matT[e];
      mstart = 1;
    }
    for (int m = mstart; m < 16; ++m) {
      v8f c = {};
#pragma unroll
      for (int k2 = 0; k2 < 4; ++k2) {
        // A[mm,k] = env[mm][4*k2+k]; lane holds row mm=col (8B-aligned float2)
        v2f a = *(const v2f*)&ev[col * 16 + k2 * 4 + dsel];
        v2f b;                            // B[k,j] = mat_t[4*k2+k][j]
        b.x = matT[m * E + (k2 * 4 + dsel) * 16 + col];
        b.y = matT[m * E + (k2 * 4 + dsel + 1) * 16 + col];
        c = __builtin_amdgcn_wmma_f32_16x16x4_f32(false, a, false, b,
                                                  (short)0, c, false, false);
      }
      // env = sigmoid(env @ mat_t), written back in row-major [mm*16+j]
#pragma unroll
      for (int v = 0; v < 8; ++v) {
        float s = 1.0f / (1.0f + __expf(-c[v]));
        ev[(mrow + v) * 16 + col] = s;
      }
    }
  }

  // ---- output: trace over the bond dims ----
  if (lane == 0) {
    float tr = 0.0f;
#pragma unroll
    for (int i = 0; i < CHI; ++i) tr += ev[i * 16 + i];
    out[p] = tr;
  }
}

extern "C" void kernel_launch(void* const* d_in, const int* in_sizes, int n_in,
                              void* d_out, int out_size, void* d_ws, size_t ws_size,
                              hipStream_t stream) {
  const float* x   = (const float*)d_in[0];   // (128, 64) f32
  const float* mps = (const float*)d_in[1];   // (64, 128, 32, 16, 16) f32
  float* out = (float*)d_out;                 // (128, 32) f32 = 4096
  (void)in_sizes; (void)n_in; (void)out_size; (void)d_ws; (void)ws_size;

  dim3 grid(L * O);   // 4096 workgroups, one (l,o) chain each
  dim3 block(32);     // one wave32
  tensornet_mps_scan<<<grid, block, 0, stream>>>(x, mps, out);
}